// Sampler_41669772705843
// MI455X (gfx1250) — compile-verified
//
#include <hip/hip_runtime.h>
#include <stdint.h>

// One workgroup (1024 threads = 32 wave32 waves) per batch row.
// Sort-free sampler: exact 32-bit-key radix select for top-k (T1),
// weighted radix select for top-p (T2), min-p as a value threshold,
// then Gumbel-style argmax of exp(v-m)/noise over the kept set.
// Streaming passes use float4 (global_load_b128) for bandwidth.

#define NTH  1024
#define NB1  2048          // level-1/2 bins (11 bits)
#define NB3  1024          // level-3 bins (10 bits)
#define IMAXI 0x7FFFFFFF

// Monotonic total-order key: ascending key <=> ascending float.
__device__ __forceinline__ uint32_t f2key(float f) {
  uint32_t u = __float_as_uint(f);
  return (u & 0x80000000u) ? ~u : (u | 0x80000000u);
}

// Inclusive suffix sums of a[0..n-1], executed by ONE wave (32 lanes),
// wave-synchronous shuffles, no barriers needed inside.
template <typename T>
__device__ __forceinline__ void suffix_scan_wave(T* a, int n) {
  const int lane = threadIdx.x & 31;
  T carry = (T)0;
  for (int base = n - 32; base >= 0; base -= 32) {
    T v = a[base + lane];
#pragma unroll
    for (int off = 1; off < 32; off <<= 1) {
      T o = __shfl(v, lane + off, 32);
      if (lane + off < 32) v = v + o;
    }
    T tot = __shfl(v, 0, 32);
    a[base + lane] = v + carry;
    carry = carry + tot;
  }
}

__global__ __launch_bounds__(NTH)
void sampler_kernel(const float* __restrict__ logits,
                    const float* __restrict__ temps,
                    const int*   __restrict__ topks,
                    const float* __restrict__ topps,
                    const float* __restrict__ minps,
                    const float* __restrict__ noise,
                    float* __restrict__ out, int V)
{
  const int r    = blockIdx.x;
  const int tid  = threadIdx.x;
  const int lane = tid & 31;
  const int wv   = tid >> 5;

  const float* __restrict__ row  = logits + (size_t)r * (size_t)V;
  const float* __restrict__ nrow = noise  + (size_t)r * (size_t)V;
  const float4* __restrict__ row4  = (const float4*)row;
  const float4* __restrict__ nrow4 = (const float4*)nrow;
  const int V4 = ((V & 3) == 0) ? (V >> 2) : 0;   // vector path only if divisible
  const int tail0 = V4 << 2;

  __shared__ uint32_t s_c1[NB1];  __shared__ float s_w1[NB1];
  __shared__ uint32_t s_c2[NB1];  __shared__ float s_w2[NB1];
  __shared__ uint32_t s_c3[NB3];  __shared__ float s_w3[NB3];
  __shared__ float s_pm[32];      __shared__ int   s_pi[32];
  __shared__ float s_stage[NTH];  // async-to-LDS staging target

  __shared__ float sh_temp, sh_topp, sh_minp, sh_m, sh_pZ;
  __shared__ int   sh_k, sh_b1, sh_b2, sh_b3, sh_b1p, sh_b2p, sh_b3p;
  __shared__ uint32_t sh_r1, sh_r2, sh_rd, sh_T1, sh_T2, sh_keymax;

  // ---- CDNA5 async path: stage first row tile into LDS via ASYNCcnt ----
  {
    uint32_t la = (uint32_t)(uintptr_t)&s_stage[tid];        // low 32 bits == LDS offset
    const float* ga = row + (tid < V ? tid : 0);
    asm volatile("global_load_async_to_lds_b32 %0, %1, off"
                 :: "v"(la), "v"(ga) : "memory");
    asm volatile("s_wait_asynccnt 0" ::: "memory");
  }

  if (tid == 0) {
    sh_temp = temps[r];
    sh_topp = topps[r];
    sh_minp = minps[r];
    int k = topks[r];
    if (k < 1) k = 1;
    if (k > V) k = V;
    sh_k = k;
  }
  for (int i = tid; i < NB1; i += NTH) { s_c1[i] = 0u; s_w1[i] = 0.0f; s_c2[i] = 0u; }
  __syncthreads();

  const float temp = sh_temp;

  // ================= P0: row max + level-1 count histogram =================
  float lm = -INFINITY;
  {
    auto p0 = [&](float l) {
      float v = l / temp;
      lm = fmaxf(lm, v);
      atomicAdd(&s_c1[f2key(v) >> 21], 1u);
    };
    for (int i = tid; i < V4; i += NTH) {
      if (i + 2 * NTH < V4) __builtin_prefetch(&row4[i + 2 * NTH], 0, 0);
      float4 x = row4[i];
      p0(x.x); p0(x.y); p0(x.z); p0(x.w);
    }
    for (int i = tail0 + tid; i < V; i += NTH) p0(row[i]);
  }
#pragma unroll
  for (int off = 16; off > 0; off >>= 1) lm = fmaxf(lm, __shfl_down(lm, off, 32));
  if (lane == 0) s_pm[wv] = lm;
  __syncthreads();
  if (wv == 0) {
    float x = s_pm[lane];
#pragma unroll
    for (int off = 16; off > 0; off >>= 1) x = fmaxf(x, __shfl_down(x, off, 32));
    if (lane == 0) { sh_m = x; sh_keymax = f2key(x); }
  }
  __syncthreads();
  if (wv == 0) suffix_scan_wave(s_c1, NB1);       // s_c1 -> suffix counts
  __syncthreads();
  { uint32_t k = (uint32_t)sh_k;                  // find L1 bin of k-th largest
    for (int i = tid; i < NB1; i += NTH) {
      uint32_t Si = s_c1[i], Sn = (i + 1 < NB1) ? s_c1[i + 1] : 0u;
      if (Si >= k && Sn < k) { sh_b1 = i; sh_r1 = k - Sn; }
    } }
  __syncthreads();

  const float m  = sh_m;
  const int   b1 = sh_b1;

  // ======== P1: L1 weight histogram + L2 counts (restricted to b1) ========
  {
    auto p1 = [&](float l) {
      float v = l / temp;
      uint32_t key = f2key(v);
      float w = expf(v - m);
      atomicAdd(&s_w1[key >> 21], w);
      if ((int)(key >> 21) == b1) atomicAdd(&s_c2[(key >> 10) & 0x7FFu], 1u);
    };
    for (int i = tid; i < V4; i += NTH) {
      float4 x = row4[i];
      p1(x.x); p1(x.y); p1(x.z); p1(x.w);
    }
    for (int i = tail0 + tid; i < V; i += NTH) p1(row[i]);
  }
  __syncthreads();
  if (wv == 0)      suffix_scan_wave(s_c2, NB1);  // parallel scans, disjoint arrays
  else if (wv == 1) suffix_scan_wave(s_w1, NB1);
  __syncthreads();
  { uint32_t k = sh_r1;
    for (int i = tid; i < NB1; i += NTH) {
      uint32_t Si = s_c2[i], Sn = (i + 1 < NB1) ? s_c2[i + 1] : 0u;
      if (Si >= k && Sn < k) { sh_b2 = i; sh_r2 = k - Sn; }
    } }
  for (int i = tid; i < NB1; i += NTH) s_w2[i] = 0.0f;
  for (int i = tid; i < NB3; i += NTH) { s_c3[i] = 0u; s_w3[i] = 0.0f; }
  __syncthreads();

  const int b2 = sh_b2;

  // ==== P2: L2 weights (restr. b1) + L3 count/weight (restr. b1,b2) ====
  {
    auto p2 = [&](float l) {
      float v = l / temp;
      uint32_t key = f2key(v);
      if ((int)(key >> 21) == b1) {
        float w = expf(v - m);
        int l2 = (int)((key >> 10) & 0x7FFu);
        atomicAdd(&s_w2[l2], w);
        if (l2 == b2) {
          int l3 = (int)(key & 0x3FFu);
          atomicAdd(&s_c3[l3], 1u);
          atomicAdd(&s_w3[l3], w);
        }
      }
    };
    for (int i = tid; i < V4; i += NTH) {
      float4 x = row4[i];
      p2(x.x); p2(x.y); p2(x.z); p2(x.w);
    }
    for (int i = tail0 + tid; i < V; i += NTH) p2(row[i]);
  }
  __syncthreads();
  if (wv == 0)      suffix_scan_wave(s_c3, NB3);
  else if (wv == 1) suffix_scan_wave(s_w2, NB1);
  else if (wv == 2) suffix_scan_wave(s_w3, NB3);
  __syncthreads();
  { uint32_t k = sh_r2;
    for (int i = tid; i < NB3; i += NTH) {
      uint32_t Si = s_c3[i], Sn = (i + 1 < NB3) ? s_c3[i + 1] : 0u;
      if (Si >= k && Sn < k) { sh_b3 = i; sh_rd = k - Sn; }
    } }
  __syncthreads();
  if (tid == 0) {
    int b3 = sh_b3;
    float SW1 = (b1 + 1 < NB1) ? s_w1[b1 + 1] : 0.0f;   // weight strictly above b1
    float SW2 = (b2 + 1 < NB1) ? s_w2[b2 + 1] : 0.0f;   // above b2 within b1
    float Z   = SW1 + SW2 + s_w3[b3];                   // suffix incl. T1's own key-bin
    sh_pZ = sh_topp * Z;                                // top-p mass threshold
    sh_T1 = ((uint32_t)b1 << 21) | ((uint32_t)b2 << 10) | (uint32_t)b3;
    sh_T2 = 0xFFFFFFFFu;                                // default: keep only the max
    sh_b1p = IMAXI;
  }
  __syncthreads();

  // ---- top-p: level-1 search for lowest nonempty bin with W_above < pZ ----
  { float pZ = sh_pZ;
    for (int i = tid; i < NB1; i += NTH) {
      float wex = (i + 1 < NB1) ? s_w1[i + 1] : 0.0f;
      uint32_t c = s_c1[i] - ((i + 1 < NB1) ? s_c1[i + 1] : 0u);
      if (wex < pZ && c > 0u) atomicMin(&sh_b1p, i);
    } }
  __syncthreads();

  const int b1p = sh_b1p;                 // block-uniform
  if (b1p != IMAXI) {
    // ---- P3: rebuild L2 count/weight restricted to b1p ----
    for (int i = tid; i < NB1; i += NTH) { s_c2[i] = 0u; s_w2[i] = 0.0f; }
    __syncthreads();
    {
      auto p3 = [&](float l) {
        float v = l / temp;
        uint32_t key = f2key(v);
        if ((int)(key >> 21) == b1p) {
          float w = expf(v - m);
          int l2 = (int)((key >> 10) & 0x7FFu);
          atomicAdd(&s_c2[l2], 1u);
          atomicAdd(&s_w2[l2], w);
        }
      };
      for (int i = tid; i < V4; i += NTH) {
        float4 x = row4[i];
        p3(x.x); p3(x.y); p3(x.z); p3(x.w);
      }
      for (int i = tail0 + tid; i < V; i += NTH) p3(row[i]);
    }
    __syncthreads();
    if (wv == 0) suffix_scan_wave(s_w2, NB1);   // counts stay raw (nonempty test)
    if (tid == 0) sh_b2p = IMAXI;
    __syncthreads();
    { float pZ = sh_pZ;
      float base1 = (b1p + 1 < NB1) ? s_w1[b1p + 1] : 0.0f;
      for (int i = tid; i < NB1; i += NTH) {
        float wex = base1 + ((i + 1 < NB1) ? s_w2[i + 1] : 0.0f);
        if (wex < pZ && s_c2[i] > 0u) atomicMin(&sh_b2p, i);
      } }
    __syncthreads();
    const int b2p = sh_b2p;
    if (b2p != IMAXI) {
      // ---- P4: L3 restricted to (b1p, b2p) -> exact T2 key ----
      for (int i = tid; i < NB3; i += NTH) { s_c3[i] = 0u; s_w3[i] = 0.0f; }
      __syncthreads();
      {
        auto p4 = [&](float l) {
          float v = l / temp;
          uint32_t key = f2key(v);
          if ((int)(key >> 21) == b1p && (int)((key >> 10) & 0x7FFu) == b2p) {
            float w = expf(v - m);
            int l3 = (int)(key & 0x3FFu);
            atomicAdd(&s_c3[l3], 1u);
            atomicAdd(&s_w3[l3], w);
          }
        };
        for (int i = tid; i < V4; i += NTH) {
          float4 x = row4[i];
          p4(x.x); p4(x.y); p4(x.z); p4(x.w);
        }
        for (int i = tail0 + tid; i < V; i += NTH) p4(row[i]);
      }
      __syncthreads();
      if (wv == 0) suffix_scan_wave(s_w3, NB3);
      if (tid == 0) sh_b3p = IMAXI;
      __syncthreads();
      { float pZ = sh_pZ;
        float base2 = ((b1p + 1 < NB1) ? s_w1[b1p + 1] : 0.0f)
                    + ((b2p + 1 < NB1) ? s_w2[b2p + 1] : 0.0f);
        for (int i = tid; i < NB3; i += NTH) {
          float wex = base2 + ((i + 1 < NB3) ? s_w3[i + 1] : 0.0f);
          if (wex < pZ && s_c3[i] > 0u) atomicMin(&sh_b3p, i);
        } }
      __syncthreads();
      if (tid == 0 && sh_b3p != IMAXI)
        sh_T2 = ((uint32_t)b1p << 21) | ((uint32_t)b2p << 10) | (uint32_t)sh_b3p;
    }
  }
  __syncthreads();

  // ============ PF: masked Gumbel-style argmax over kept set ============
  const uint32_t T1 = sh_T1, T2 = sh_T2, kmax = sh_keymax;
  const float minp = sh_minp;

  float bs = -1.0f; int bi = IMAXI;
  {
    auto pf = [&](float l, float nzraw, int idx) {
      float v = l / temp;
      uint32_t key = f2key(v);
      bool keep; float w;
      if (key == kmax) { keep = true; w = expf(v - m); }    // max never masked
      else {
        keep = (key >= T1) && (key >= T2);
        w = 0.0f;
        if (keep) {
          w = expf(v - m);
          if (minp > 0.0f && w < minp) keep = false;        // min-p threshold
        }
      }
      if (keep) {
        float nz = fmaxf(nzraw, 1e-10f);
        float sc = w / nz;
        if (sc > bs || (sc == bs && idx < bi)) { bs = sc; bi = idx; }
      }
    };
    for (int i = tid; i < V4; i += NTH) {
      float4 x = row4[i];
      float4 nz = nrow4[i];
      int base = i << 2;
      pf(x.x, nz.x, base); pf(x.y, nz.y, base + 1);
      pf(x.z, nz.z, base + 2); pf(x.w, nz.w, base + 3);
    }
    for (int i = tail0 + tid; i < V; i += NTH) pf(row[i], nrow[i], i);
  }
#pragma unroll
  for (int off = 16; off > 0; off >>= 1) {
    float os = __shfl_down(bs, off, 32);
    int   oi = __shfl_down(bi, off, 32);
    if (os > bs || (os == bs && oi < bi)) { bs = os; bi = oi; }
  }
  if (lane == 0) { s_pm[wv] = bs; s_pi[wv] = bi; }
  __syncthreads();
  if (wv == 0) {
    float xs = s_pm[lane]; int xi = s_pi[lane];
#pragma unroll
    for (int off = 16; off > 0; off >>= 1) {
      float os = __shfl_down(xs, off, 32);
      int   oi = __shfl_down(xi, off, 32);
      if (os > xs || (os == xs && oi < xi)) { xs = os; xi = oi; }
    }
    if (lane == 0) out[r] = (float)xi;
  }
}

extern "C" void kernel_launch(void* const* d_in, const int* in_sizes, int n_in,
                              void* d_out, int out_size, void* d_ws, size_t ws_size,
                              hipStream_t stream) {
  (void)n_in; (void)d_ws; (void)ws_size; (void)out_size;
  const float* logits = (const float*)d_in[0];
  const float* temps  = (const float*)d_in[1];
  const int*   topks  = (const int*)d_in[2];
  const float* topps  = (const float*)d_in[3];
  const float* minps  = (const float*)d_in[4];
  const float* noise  = (const float*)d_in[5];
  const int B = in_sizes[1];
  const int V = in_sizes[0] / B;
  sampler_kernel<<<dim3(B), dim3(NTH), 0, stream>>>(
      logits, temps, topks, topps, minps, noise, (float*)d_out, V);
}